// GroupedEmbeddingBag_49864570306747
// MI455X (gfx1250) — compile-verified
//
#include <hip/hip_runtime.h>
#include <stdint.h>

// Problem constants (fixed by the reference).
#define T_TABLES 8
#define N_ROWS   200000
#define D_DIM    128
#define B_BAGS   8192
#define L_IDX    163840

#define ROW_BYTES      (D_DIM * 4)                   // 512 B per embedding row
#define ROWS_PER_CHUNK 8                             // 32-bit gather indices -> max 8 rows per TDM op
#define CHUNK_BYTES    (ROWS_PER_CHUNK * ROW_BYTES)  // 4 KB
#define WAVES_PER_BLOCK 8
#define LDS_PER_WAVE   (2 * CHUNK_BYTES)             // double buffer: 8 KB
#define LDS_TOTAL      (WAVES_PER_BLOCK * LDS_PER_WAVE) // 64 KB per block

typedef unsigned int u32x4 __attribute__((ext_vector_type(4)));
typedef int          i32x4 __attribute__((ext_vector_type(4)));
typedef int          i32x8 __attribute__((ext_vector_type(8)));

// Issue one TDM gather: fetch n rows (row indices r0..r7) of 128 f32 each from
// the embedding table at table_base into LDS at byte offset lds_addr.
__device__ __forceinline__ void tdm_gather_rows(uint64_t table_base,
                                                uint32_t lds_addr, int n,
                                                int r0, int r1, int r2, int r3,
                                                int r4, int r5, int r6, int r7) {
  u32x4 g0;
  // count=1, is_restore=0, gather_index_size=1 (32-bit), gather_mode=1, type=2
  g0[0] = 0xC0000001u;
  g0[1] = lds_addr;                                   // LDS byte address
  g0[2] = (uint32_t)(table_base & 0xFFFFFFFFu);       // global_addr[31:0]
  g0[3] = ((uint32_t)(table_base >> 32) & 0x01FFFFFFu) | (2u << 30); // [56:32] | type=2

  i32x8 g1;
  g1[0] = (int)(2u << 16);                            // wg_mask=0, data_size=2 (4B)
  g1[1] = (int)((uint32_t)D_DIM << 16);               // tensor_dim0[15:0]=128
  g1[2] = (int)(((uint32_t)N_ROWS & 0xFFFFu) << 16);  // tensor_dim1[15:0]
  g1[3] = (int)(((uint32_t)N_ROWS >> 16) |            // tensor_dim1[31:16]
                ((uint32_t)D_DIM << 16));             // tile_dim0=128
  g1[4] = n;                                          // tile_dim1 = #valid indices
  g1[5] = D_DIM;                                      // tensor_dim0_stride = 128 elems
  g1[6] = 0;                                          // dim1_stride ignored in gather mode
  g1[7] = 0;

  i32x4 g2; g2[0] = r0; g2[1] = r1; g2[2] = r2; g2[3] = r3; // row_index_0..3
  i32x4 g3; g3[0] = r4; g3[1] = r5; g3[2] = r6; g3[3] = r7; // row_index_4..7

  i32x8 g4 = {0, 0, 0, 0, 0, 0, 0, 0};                // unused trailing group (VADDR4 slot)

  __builtin_amdgcn_tensor_load_to_lds(g0, g1, g2, g3, g4, 0);
}

__device__ __forceinline__ void accum_rows(const char* bufp, int lane, int nrows,
                                           float& ax, float& ay, float& az, float& aw) {
  for (int j = 0; j < nrows; ++j) {
    const float4 r = *(const float4*)(bufp + j * ROW_BYTES + lane * 16);
    ax += r.x; ay += r.y; az += r.z; aw += r.w;
  }
}

extern "C" __global__ void __launch_bounds__(WAVES_PER_BLOCK * 32)
emb_pool_tdm(const float* __restrict__ weights, const int* __restrict__ values,
             const int* __restrict__ offsets, float* __restrict__ out) {
  extern __shared__ char smem[];

  const int lane      = threadIdx.x & 31;
  const int waveInBlk = threadIdx.x >> 5;
  const int w = blockIdx.x * WAVES_PER_BLOCK + waveInBlk; // 0 .. T*B-1
  const int t = w >> 13;            // / B_BAGS
  const int b = w & (B_BAGS - 1);

  // Bag bounds (wave-uniform -> force into SGPRs).
  const int* offp = offsets + t * (B_BAGS + 1) + b;
  int start = __builtin_amdgcn_readfirstlane(offp[0]);
  int end   = __builtin_amdgcn_readfirstlane(offp[1]);
  const int num = end - start;

  const int st = __builtin_amdgcn_readfirstlane(t);
  const uint64_t table_base =
      (uint64_t)(uintptr_t)weights + (uint64_t)st * ((uint64_t)N_ROWS * D_DIM * 4);
  const int* idxp = values + (size_t)t * L_IDX + start;

  const int swave = __builtin_amdgcn_readfirstlane(waveInBlk);
  const uint32_t ldsWave =
      __builtin_amdgcn_groupstaticsize() + (uint32_t)swave * LDS_PER_WAVE;
  const char* bufbase = smem + (size_t)waveInBlk * LDS_PER_WAVE;

  float ax = 0.f, ay = 0.f, az = 0.f, aw = 0.f;

  const int chunks = (num + ROWS_PER_CHUNK - 1) / ROWS_PER_CHUNK;
  if (chunks > 0) {
    // ---- prologue: issue chunk 0 into buffer 0 ----
    {
      const int nn = num < ROWS_PER_CHUNK ? num : ROWS_PER_CHUNK;
      int v = 0;
      if (lane < nn) v = idxp[lane];
      tdm_gather_rows(table_base, ldsWave, nn,
                      __builtin_amdgcn_readlane(v, 0), __builtin_amdgcn_readlane(v, 1),
                      __builtin_amdgcn_readlane(v, 2), __builtin_amdgcn_readlane(v, 3),
                      __builtin_amdgcn_readlane(v, 4), __builtin_amdgcn_readlane(v, 5),
                      __builtin_amdgcn_readlane(v, 6), __builtin_amdgcn_readlane(v, 7));
    }
    // ---- steady state: issue chunk c, drain chunk c-1 ----
    for (int c = 1; c < chunks; ++c) {
      const int rem = num - c * ROWS_PER_CHUNK;
      const int nn  = rem < ROWS_PER_CHUNK ? rem : ROWS_PER_CHUNK;
      int v = 0;
      if (lane < nn) v = idxp[c * ROWS_PER_CHUNK + lane];
      const int r0 = __builtin_amdgcn_readlane(v, 0), r1 = __builtin_amdgcn_readlane(v, 1);
      const int r2 = __builtin_amdgcn_readlane(v, 2), r3 = __builtin_amdgcn_readlane(v, 3);
      const int r4 = __builtin_amdgcn_readlane(v, 4), r5 = __builtin_amdgcn_readlane(v, 5);
      const int r6 = __builtin_amdgcn_readlane(v, 6), r7 = __builtin_amdgcn_readlane(v, 7);
      // WAR guard: our ds_loads from this buffer (previous iteration) must be done.
      asm volatile("s_wait_dscnt 0" ::: "memory");
      tdm_gather_rows(table_base, ldsWave + (uint32_t)(c & 1) * CHUNK_BYTES, nn,
                      r0, r1, r2, r3, r4, r5, r6, r7);
      // Allow 1 outstanding tensor op -> chunk c-1 has fully landed in LDS.
      __builtin_amdgcn_s_wait_tensorcnt(1);
      accum_rows(bufbase + ((c - 1) & 1) * CHUNK_BYTES, lane, ROWS_PER_CHUNK,
                 ax, ay, az, aw);
    }
    // ---- epilogue: drain last (possibly partial) chunk ----
    __builtin_amdgcn_s_wait_tensorcnt(0);
    const int lastn = num - (chunks - 1) * ROWS_PER_CHUNK;
    accum_rows(bufbase + ((chunks - 1) & 1) * CHUNK_BYTES, lane, lastn,
               ax, ay, az, aw);
  }

  // out[b, t*D + lane*4 .. +3]  (512B coalesced store per wave)
  float* op = out + (size_t)b * (T_TABLES * D_DIM) + t * D_DIM + lane * 4;
  float4 acc; acc.x = ax; acc.y = ay; acc.z = az; acc.w = aw;
  *(float4*)op = acc;
}

extern "C" void kernel_launch(void* const* d_in, const int* in_sizes, int n_in,
                              void* d_out, int out_size, void* d_ws, size_t ws_size,
                              hipStream_t stream) {
  (void)in_sizes; (void)n_in; (void)out_size; (void)d_ws; (void)ws_size;
  const float* weights = (const float*)d_in[0];
  const int*   values  = (const int*)d_in[1];
  const int*   offsets = (const int*)d_in[2];
  float*       out     = (float*)d_out;

  dim3 grid((T_TABLES * B_BAGS) / WAVES_PER_BLOCK); // 8192 blocks
  dim3 block(WAVES_PER_BLOCK * 32);                 // 8 wave32s
  hipLaunchKernelGGL(emb_pool_tdm, grid, block, LDS_TOTAL, stream,
                     weights, values, offsets, out);
}